// NUFFT_24111946400516
// MI455X (gfx1250) — compile-verified
//
#include <hip/hip_runtime.h>
#include <hip/hip_bf16.h>
#include <math.h>

// ---------------------------------------------------------------------------
// NUFFT forward for MI455X (gfx1250, wave32).
//   per ph:  apodize+pad -> row 640-pt DFT -> col 640-pt DFT (TDM-staged,
//   8 columns/WG) -> 6x6 Kaiser-Bessel interpolation.
//   640-pt DFT = Cooley-Tukey 16 x 40, both stages on V_WMMA_F32_16X16X4_F32.
//   All twiddles (W16, W40, W640) precomputed into L2-resident tables.
// Workspace: 32 KB tables + 64*640*640*8 B grid (~210 MB), reused per ph.
// ---------------------------------------------------------------------------

typedef __attribute__((ext_vector_type(2))) float v2f;
typedef __attribute__((ext_vector_type(4))) float v4f;
typedef __attribute__((ext_vector_type(8))) float v8f;
typedef unsigned int u32x4 __attribute__((ext_vector_type(4)));
typedef int          i32x4 __attribute__((ext_vector_type(4)));
typedef int          i32x8 __attribute__((ext_vector_type(8)));

#define PI_F     3.14159265358979323846f
#define KB_JW    6
#define KB_ALPHA 14.04f          // 2.34 * J
#define NIM      320
#define NGRID    640
#define NCH      64              // ch * z
#define NSAMP    25600

// workspace float offsets
#define WS_SC     0              // 320
#define WS_W16R   512            // 256  : W16[m*16+k]  = exp(-2pi i m k/16)
#define WS_W16I   768
#define WS_W40TR  1024           // 2304 : W40t[n*48+k] = exp(-2pi i k n/40)
#define WS_W40TI  3328
#define WS_TWR    5632           // 768  : TW[c*16+k1]  = exp(-2pi i k1 c/640)
#define WS_TWI    6400
#define WS_GRID   8192           // float2 grid starts at byte 32768

static __device__ __forceinline__ v8f wmma4(v2f a, v2f b, v8f c) {
  // D = A(16x4,f32) * B(4x16,f32) + C(16x16,f32)
  return __builtin_amdgcn_wmma_f32_16x16x4_f32(false, a, false, b, (short)0, c,
                                               false, false);
}

// Abramowitz & Stegun 9.8.1 / 9.8.2 modified Bessel I0
static __device__ __forceinline__ float i0f(float x) {
  float ax = fabsf(x);
  if (ax < 3.75f) {
    float t = x / 3.75f; t *= t;
    return 1.0f + t*(3.5156229f + t*(3.0899424f + t*(1.2067492f
         + t*(0.2659732f + t*(0.0360768f + t*0.0045813f)))));
  }
  float t = 3.75f / ax;
  return (expf(ax) / sqrtf(ax)) * (0.39894228f + t*(0.01328592f
       + t*(0.00225319f + t*(-0.00157565f + t*(0.00916281f
       + t*(-0.02057706f + t*(0.02635537f + t*(-0.01647633f
       + t*0.00392377f))))))));
}

// ---------------------------------------------------------------------------
// 640-point complex DFT, one workgroup of 96 threads (3 full wave32s).
//   x is read as  xr[n*xs] (re) / xi[n*xs] (im); slots n=640.. must be zero.
//   out written as outr[k*os] / outi[k*os].
// ---------------------------------------------------------------------------
static __device__ __forceinline__ void dft640_wmma(
    const float* __restrict__ xr, const float* __restrict__ xi, const int xs,
    float* __restrict__ tr, float* __restrict__ ti,          // LDS 16 x 48
    float* __restrict__ outr, float* __restrict__ outi, const int os,
    const float* __restrict__ w16r, const float* __restrict__ w16i,
    const float* __restrict__ w40tr, const float* __restrict__ w40ti,
    const float* __restrict__ twr,  const float* __restrict__ twi)
{
  const int tid  = threadIdx.x;
  const int wv   = tid >> 5;      // 0..2 : 16-wide column block
  const int lane = tid & 31;
  const int hf   = lane >> 4;     // half-wave select
  const int l16  = lane & 15;

  const int  col = 16 * wv + l16; // n2 (stage A) / k2 (stage C)
  const bool cok = (col < 40);

  // ---------------- Stage A : Y(16x40) = W16(16x16) * X(16x40) -------------
  v8f arbr = {}, aibi = {}, imac = {};
  #pragma unroll
  for (int c = 0; c < 4; ++c) {
    const int k0 = 4 * c + 2 * hf;                 // n1 for .x ; +1 for .y
    v2f Ar = *(const v2f*)(w16r + l16 * 16 + k0);  // global_load_b64 (cached)
    v2f Ai = *(const v2f*)(w16i + l16 * 16 + k0);
    // unconditional LDS loads; lanes with col>=40 read the zeroed pad slot
    const int ia = cok ? (40 * k0 + col)      : 640;
    const int ib = cok ? (40 * k0 + 40 + col) : 640;
    v2f Br, Bi;
    Br.x = xr[ia * xs]; Br.y = xr[ib * xs];
    Bi.x = xi[ia * xs]; Bi.y = xi[ib * xs];
    arbr = wmma4(Ar, Br, arbr);
    aibi = wmma4(Ai, Bi, aibi);
    imac = wmma4(Ar, Bi, imac);
    imac = wmma4(Ai, Br, imac);
  }

  // twiddle W640^{k1*col}; this lane's k1 set is contiguous: 8*hf .. 8*hf+7
  float tc[8], ts[8];
  *(v4f*)&tc[0] = *(const v4f*)(twr + col * 16 + 8 * hf);
  *(v4f*)&tc[4] = *(const v4f*)(twr + col * 16 + 8 * hf + 4);
  *(v4f*)&ts[0] = *(const v4f*)(twi + col * 16 + 8 * hf);
  *(v4f*)&ts[4] = *(const v4f*)(twi + col * 16 + 8 * hf + 4);
  #pragma unroll
  for (int v = 0; v < 8; ++v) {
    const int k1 = v + 8 * hf;                 // C/D layout: VGPR v -> M=v/v+8
    float re = arbr[v] - aibi[v];
    float im = imac[v];
    tr[k1 * 48 + col] = cok ? (re * tc[v] - im * ts[v]) : 0.0f;
    ti[k1 * 48 + col] = cok ? (re * ts[v] + im * tc[v]) : 0.0f;
  }
  __syncthreads();

  // ---------------- Stage C : Out(16x40) = T(16x48) * W40(48x40) -----------
  v8f r1 = {}, r2 = {}, io = {};
  #pragma unroll
  for (int c = 0; c < 12; ++c) {
    const int n0 = 4 * c + 2 * hf;                  // n2 (K) for .x ; +1 for .y
    v2f Ar = *(const v2f*)(tr + l16 * 48 + n0);     // ds_load_b64
    v2f Ai = *(const v2f*)(ti + l16 * 48 + n0);
    v2f Br = *(const v2f*)(w40tr + col * 48 + n0);  // global_load_b64 (cached)
    v2f Bi = *(const v2f*)(w40ti + col * 48 + n0);
    r1 = wmma4(Ar, Br, r1);
    r2 = wmma4(Ai, Bi, r2);
    io = wmma4(Ar, Bi, io);
    io = wmma4(Ai, Br, io);
  }
  __syncthreads();                                  // done reading tr/ti

  #pragma unroll
  for (int v = 0; v < 8; ++v) {
    const int k1 = v + 8 * hf;
    if (cok) {
      const int k = k1 + 16 * col;                  // k = k1 + 16*k2
      outr[k * os] = r1[v] - r2[v];
      outi[k * os] = io[v];
    }
  }
  __syncthreads();
}

// ---------------------------------------------------------------------------
// Kernels
// ---------------------------------------------------------------------------
__global__ void init_tables_kernel(float* __restrict__ ws) {
  const int i = blockIdx.x * blockDim.x + threadIdx.x;     // 0..2303
  const float I0A = i0f(KB_ALPHA);
  if (i < NIM) {
    float om = ((float)i - 159.5f) / (float)NGRID;
    float a  = PI_F * (float)KB_JW * om;
    float t  = sqrtf(KB_ALPHA * KB_ALPHA - a * a);         // always > 0 here
    ws[WS_SC + i] = t * I0A / ((float)KB_JW * sinhf(t));   // 1 / kb_ft
  }
  if (i < 256) {
    int m = i >> 4, k = i & 15;
    float s, c;
    __sincosf(-(2.0f * PI_F / 16.0f) * (float)(m * k), &s, &c);
    ws[WS_W16R + i] = c;  ws[WS_W16I + i] = s;
  }
  if (i < 2304) {
    int n = i / 48, k = i % 48;                            // B[k][n] = W40^{k*n}
    float s, c;
    __sincosf(-(2.0f * PI_F / 40.0f) * (float)(k * n), &s, &c);
    ws[WS_W40TR + i] = c;  ws[WS_W40TI + i] = s;
  }
  if (i < 768) {
    int cc = i / 16, k1 = i % 16;
    float s, c;
    __sincosf(-(2.0f * PI_F / 640.0f) * (float)(k1 * cc), &s, &c);
    ws[WS_TWR + i] = c;  ws[WS_TWI + i] = s;
  }
}

__global__ void __launch_bounds__(96) rowfft_kernel(
    const float* __restrict__ imr, const float* __restrict__ imi,
    const float* __restrict__ ws, float2* __restrict__ grid)
{
  __shared__ float xr[656], xi[656], tr[768], ti[768], outr[640], outi[640];
  const int row = blockIdx.x;                  // 0..319 (rows >=320 are zero)
  const int ch  = blockIdx.y;                  // 0..63
  const size_t ioff = ((size_t)ch * NIM + row) * NIM;
  const float* sc = ws + WS_SC;
  const float sy = sc[row];
  for (int x = threadIdx.x; x < 656; x += 96) {
    float wr = 0.0f, wi = 0.0f;
    if (x < NIM) {
      float w = sy * sc[x];
      wr = imr[ioff + x] * w;
      wi = imi[ioff + x] * w;
    }
    xr[x] = wr; xi[x] = wi;
  }
  __syncthreads();
  dft640_wmma(xr, xi, 1, tr, ti, outr, outi, 1,
              ws + WS_W16R, ws + WS_W16I, ws + WS_W40TR, ws + WS_W40TI,
              ws + WS_TWR, ws + WS_TWI);
  const size_t goff = ((size_t)ch * NGRID + row) * NGRID;
  const float nrm = 1.0f / 640.0f;             // fft2 ortho norm, applied once
  for (int x = threadIdx.x; x < NGRID; x += 96)
    grid[goff + x] = make_float2(outr[x] * nrm, outi[x] * nrm);
}

__global__ void __launch_bounds__(96) colfft_kernel(
    float2* __restrict__ grid, const float* __restrict__ ws)
{
  __shared__ float2 xc[656][8];                // 8 columns, rows 320.. zeroed
  __shared__ float  tr[768], ti[768];
  __shared__ float2 ob[640][8];                // output staging (coalesced store)
  const int x0 = blockIdx.x * 8;               // first of 8 columns
  const int ch = blockIdx.y;
  const size_t base = (size_t)ch * NGRID * NGRID;

  // zero rows 320..655 (zero-pad region + WMMA pad slots); disjoint from TDM
  for (int p = threadIdx.x; p < (656 - 320) * 8; p += 96) {
    xc[320 + (p >> 3)][p & 7] = make_float2(0.0f, 0.0f);
  }

#if __has_builtin(__builtin_amdgcn_tensor_load_to_lds)
  // Tensor Data Mover: 2-D tile, 16 floats (8 float2 cols) x 320 rows,
  // row stride 1280 floats, into compacted LDS tile xc[0..319][0..7].
  if (threadIdx.x < 32) {                      // wave 0 issues the DMA
    unsigned long long ga = (unsigned long long)(uintptr_t)(grid + base + x0);
    unsigned int lds = (unsigned int)(uintptr_t)(&xc[0][0]);
    u32x4 g0;
    g0[0] = 1u;                                          // count=1
    g0[1] = lds;                                         // lds_addr (bytes)
    g0[2] = (unsigned int)(ga & 0xffffffffull);          // global_addr lo
    g0[3] = (unsigned int)((ga >> 32) & 0x01ffffffull)   // global_addr hi
          | (2u << 30);                                  // type=2 (image)
    i32x8 g1;
    g1[0] = (int)(2u << 16);          // workgroup_mask=0, data_size=2 (4B)
    g1[1] = (int)(1280u << 16);       // tensor_dim0[15:0]=1280 in bits[63:48]
    g1[2] = (int)(320u << 16);        // tensor_dim0 hi=0 | tensor_dim1=320
    g1[3] = (int)(16u << 16);         // tensor_dim1 hi=0 | tile_dim0=16
    g1[4] = 320;                      // tile_dim1=320, tile_dim2=0
    g1[5] = 1280;                     // tensor_dim0_stride (floats)
    g1[6] = 0; g1[7] = 0;             // dim1 stride unused for 2-D tile
    i32x4 gz4 = {0, 0, 0, 0};
    i32x8 gz8 = {0, 0, 0, 0, 0, 0, 0, 0};
    // clang-23 / therock-10.0 arity: (g0, g1, g2, g3, g4, cpol)
    __builtin_amdgcn_tensor_load_to_lds(g0, g1, gz4, gz4, gz8, 0);
    __builtin_amdgcn_s_wait_tensorcnt(0);
  }
#else
  for (int p = threadIdx.x; p < NIM * 8; p += 96) {
    int y = p >> 3, c = p & 7;
    xc[y][c] = grid[base + (size_t)y * NGRID + x0 + c];
  }
#endif
  __syncthreads();

  for (int l = 0; l < 8; ++l) {
    dft640_wmma(&xc[0][l].x, &xc[0][l].y, 16, tr, ti,
                &ob[0][l].x, &ob[0][l].y, 16,
                ws + WS_W16R, ws + WS_W16I, ws + WS_W40TR, ws + WS_W40TI,
                ws + WS_TWR, ws + WS_TWI);
  }

  for (int p = threadIdx.x; p < NGRID * 8; p += 96) {
    int y = p >> 3, c = p & 7;
    grid[base + (size_t)y * NGRID + x0 + c] = ob[y][c];
  }
}

__global__ void interp_kernel(const float2* __restrict__ grid,
                              const float* __restrict__ traj,
                              float* __restrict__ out)
{
  const int s = blockIdx.x * blockDim.x + threadIdx.x;
  if (s >= NSAMP) return;
  const float om_y = traj[s];
  const float om_x = traj[NSAMP + s];
  const float scl  = (float)NGRID / (2.0f * PI_F);
  const float tm_y = om_y * scl, tm_x = om_x * scl;
  const float I0A  = i0f(KB_ALPHA);

  int   rowoff[6], colix[6];
  float wy[6], wx[6];
  const int oy = (int)floorf(tm_y - 3.0f);
  const int ox = (int)floorf(tm_x - 3.0f);
  #pragma unroll
  for (int j = 1; j <= 6; ++j) {
    int ky = oy + j, kx = ox + j;
    float uy = tm_y - (float)ky, ux = tm_x - (float)kx;
    float ay = 1.0f - (uy / 3.0f) * (uy / 3.0f);
    float ax = 1.0f - (ux / 3.0f) * (ux / 3.0f);
    wy[j - 1] = (ay > 0.0f) ? i0f(KB_ALPHA * sqrtf(ay)) / I0A : 0.0f;
    wx[j - 1] = (ax > 0.0f) ? i0f(KB_ALPHA * sqrtf(ax)) / I0A : 0.0f;
    int iy = ky % NGRID; if (iy < 0) iy += NGRID;
    int ix = kx % NGRID; if (ix < 0) ix += NGRID;
    rowoff[j - 1] = iy * NGRID;
    colix[j - 1]  = ix;
  }
  float pc, ps;   // exp(i*(om_y*160 + om_x*160))
  __sincosf(om_y * 160.0f + om_x * 160.0f, &ps, &pc);

  for (int c = 0; c < NCH; ++c) {
    const float2* g = grid + (size_t)c * NGRID * NGRID;
    float ar = 0.0f, ai = 0.0f;
    #pragma unroll
    for (int jy = 0; jy < 6; ++jy) {
      #pragma unroll
      for (int jx = 0; jx < 6; ++jx) {
        float w = wy[jy] * wx[jx];
        float2 v = g[rowoff[jy] + colix[jx]];
        ar += w * v.x;
        ai += w * v.y;
      }
    }
    out[((size_t)c * NSAMP + s) * 2 + 0] = ar * pc - ai * ps;
    out[((size_t)c * NSAMP + s) * 2 + 1] = ar * ps + ai * pc;
  }
}

// ---------------------------------------------------------------------------
extern "C" void kernel_launch(void* const* d_in, const int* in_sizes, int n_in,
                              void* d_out, int out_size, void* d_ws, size_t ws_size,
                              hipStream_t stream)
{
  const float* imr  = (const float*)d_in[0];   // [4,8,8,320,320]
  const float* imi  = (const float*)d_in[1];   // [4,8,8,320,320]
  const float* traj = (const float*)d_in[2];   // [4,2,25600]
  float* out = (float*)d_out;                  // [4,8,8,25600] cplx -> float pairs

  float*  ws   = (float*)d_ws;
  float2* grid = (float2*)(ws + WS_GRID);      // 64*640*640 complex (~210 MB)

  init_tables_kernel<<<9, 256, 0, stream>>>(ws);

  for (int ph = 0; ph < 4; ++ph) {
    const size_t imoff = (size_t)ph * NCH * NIM * NIM;
    rowfft_kernel<<<dim3(NIM, NCH), 96, 0, stream>>>(imr + imoff, imi + imoff,
                                                     ws, grid);
    colfft_kernel<<<dim3(NGRID / 8, NCH), 96, 0, stream>>>(grid, ws);
    interp_kernel<<<dim3((NSAMP + 255) / 256), 256, 0, stream>>>(
        grid, traj + (size_t)ph * 2 * NSAMP,
        out + (size_t)ph * NCH * NSAMP * 2);
  }
}